// Distance_Deterministic_18081812316190
// MI455X (gfx1250) — compile-verified
//
#include <hip/hip_runtime.h>
#include <hip/hip_bf16.h>
#include <math.h>

// CDNA5 (gfx1250) radius-graph with strict K-nearest cap.
// Gram matrix P·P^T computed 16x16-tile-at-a-time with V_WMMA_F32_16X16X4_F32
// (K padded 3 -> 4 with zeros), full f32 precision to match the reference.

typedef __attribute__((ext_vector_type(2))) float v2f;
typedef __attribute__((ext_vector_type(8))) float v8f;

#define KNBR 32          // max neighbors
#define CAP  128         // per-row candidate capacity (mean ~49 in cutoff)
#define CUT2 25.0f       // CUTOFF^2

static __device__ __forceinline__ unsigned long long shflxor64(unsigned long long v, int m) {
    unsigned lo = (unsigned)(v & 0xffffffffull);
    unsigned hi = (unsigned)(v >> 32);
    lo = __shfl_xor(lo, m, 32);
    hi = __shfl_xor(hi, m, 32);
    return ((unsigned long long)hi << 32) | (unsigned long long)lo;
}

__global__ __launch_bounds__(128) void radius_graph_wmma_f32(
    const float* __restrict__ pos, int n,
    int* __restrict__ ei,        // [2*E] edge_index as int32 bits
    float* __restrict__ wt,      // [E]   edge_weight
    float* __restrict__ vc,      // [3*E] edge_vec
    int E) {

    __shared__ unsigned s_cnt[16];
    __shared__ float    s_d2 [16 * CAP];
    __shared__ int      s_idx[16 * CAP];

    const int lane = threadIdx.x & 31;
    const int wv   = threadIdx.x >> 5;      // 4 waves per block
    const int hi   = lane >> 4;             // lanes 16..31 carry K=2,3
    const int i0   = blockIdx.x * 16;       // 16 rows per block

    if (threadIdx.x < 16) s_cnt[threadIdx.x] = 0u;
    __syncthreads();

    // ---- A operand: rows i0..i0+15, 16x4 f32 layout ----
    // lanes 0-15:  VGPR0=K0(x), VGPR1=K1(y); lanes 16-31: VGPR0=K2(z), VGPR1=K3(0)
    const int ia = i0 + (lane & 15);
    const float ax0 = pos[3 * ia + 0];
    const float ay0 = pos[3 * ia + 1];
    const float az0 = pos[3 * ia + 2];
    v2f a;
    a.x = hi ? az0 : ax0;
    a.y = hi ? 0.0f : ay0;

    // |p_row|^2 for the 8 rows this lane accumulates (C layout: M = v + 8*hi)
    float sqRow[8];
#pragma unroll
    for (int v = 0; v < 8; ++v) {
        int r = i0 + v + 8 * hi;
        float x = pos[3 * r], y = pos[3 * r + 1], z = pos[3 * r + 2];
        sqRow[v] = x * x + y * y + z * z;
    }

    const int ntiles = n >> 4;
    const int tpw    = ntiles >> 2;            // column tiles per wave
    const int t0 = wv * tpw, t1 = t0 + tpw;

    for (int t = t0; t < t1; ++t) {
        const int j0 = t << 4;
        const int ja = j0 + (lane & 15);
        const float px = pos[3 * ja], py = pos[3 * ja + 1], pz = pos[3 * ja + 2];
        const float sqj = px * px + py * py + pz * pz;

        // B operand: cols j0..j0+15, 4x16 f32 layout (mirror of A)
        v2f b;
        b.x = hi ? pz : px;
        b.y = hi ? 0.0f : py;

        if (t + 1 < t1) __builtin_prefetch(pos + 3 * (j0 + 16), 0, 0);

        v8f c = {};
#if __has_builtin(__builtin_amdgcn_wmma_f32_16x16x4_f32)
        // (neg_a, A, neg_b, B, c_mod, C, reuse_a, reuse_b)
        c = __builtin_amdgcn_wmma_f32_16x16x4_f32(
                false, a, false, b, (short)0, c, false, false);
#else
        // scalar fallback (keeps compiling; signals wmma=0 in histogram)
#pragma unroll
        for (int v = 0; v < 8; ++v) {
            int r = i0 + v + 8 * hi;
            c[v] = pos[3 * r] * px + pos[3 * r + 1] * py + pos[3 * r + 2] * pz;
        }
#endif

        // d2 = |p_i|^2 + |p_j|^2 - 2 p_i·p_j ; clamp, cutoff, exclude self
#pragma unroll
        for (int v = 0; v < 8; ++v) {
            const int rloc = v + 8 * hi;
            const int r = i0 + rloc;
            float d2 = sqRow[v] + sqj - 2.0f * c[v];
            d2 = fmaxf(d2, 0.0f);
            if (ja != r && d2 <= CUT2) {
                unsigned slot = atomicAdd(&s_cnt[rloc], 1u);
                if (slot < CAP) {
                    s_d2 [rloc * CAP + slot] = d2;
                    s_idx[rloc * CAP + slot] = ja;
                }
            }
        }
    }
    __syncthreads();

    // ---- Selection: each wave sorts/emits 4 rows (32 smallest, ties -> lower j) ----
    for (int rr = 0; rr < 4; ++rr) {
        const int rloc = wv * 4 + rr;
        const int i = i0 + rloc;
        unsigned cnt = s_cnt[rloc];
        if (cnt > CAP) cnt = CAP;
        const float pix = pos[3 * i], piy = pos[3 * i + 1], piz = pos[3 * i + 2];

        unsigned long long key[4];
#pragma unroll
        for (int p = 0; p < 4; ++p) {
            unsigned s = lane + 32u * p;
            if (s < cnt) {
                float d2v = s_d2[rloc * CAP + s];
                unsigned jv = (unsigned)s_idx[rloc * CAP + s];
                key[p] = ((unsigned long long)__float_as_uint(d2v) << 32) | jv;
            } else {
                key[p] = ~0ull;
            }
        }

        for (int k = 0; k < KNBR; ++k) {
            unsigned long long g = key[0];
#pragma unroll
            for (int p = 1; p < 4; ++p) g = (key[p] < g) ? key[p] : g;
#pragma unroll
            for (int off = 16; off > 0; off >>= 1) {
                unsigned long long o = shflxor64(g, off);
                g = (o < g) ? o : g;
            }
            if (lane == k) {
                const int e = i * KNBR + k;
                if (g != ~0ull) {
                    float d2v = __uint_as_float((unsigned)(g >> 32));
                    int j = (int)(unsigned)(g & 0xffffffffu);
                    ei[e] = j;
                    ei[E + e] = i;
                    wt[e] = sqrtf(d2v);
                    float jx = pos[3 * j], jy = pos[3 * j + 1], jz = pos[3 * j + 2];
                    vc[3 * e + 0] = jx - pix;
                    vc[3 * e + 1] = jy - piy;
                    vc[3 * e + 2] = jz - piz;
                } else {
                    ei[e] = i;
                    ei[E + e] = i;
                    wt[e] = 0.0f;
                    vc[3 * e + 0] = 0.0f;
                    vc[3 * e + 1] = 0.0f;
                    vc[3 * e + 2] = 0.0f;
                }
            }
            // consume selected key (keys unique: include j)
#pragma unroll
            for (int p = 0; p < 4; ++p)
                if (key[p] == g) key[p] = ~0ull;
        }
    }

    // ---- Appended self loops: edge N*K + i ----
    if (threadIdx.x < 16) {
        const int i = i0 + (int)threadIdx.x;
        const int e = n * KNBR + i;
        ei[e] = i;
        ei[E + e] = i;
        wt[e] = 0.0f;
        vc[3 * e + 0] = 0.0f;
        vc[3 * e + 1] = 0.0f;
        vc[3 * e + 2] = 0.0f;
    }
}

extern "C" void kernel_launch(void* const* d_in, const int* in_sizes, int n_in,
                              void* d_out, int out_size, void* d_ws, size_t ws_size,
                              hipStream_t stream) {
    (void)n_in; (void)out_size; (void)d_ws; (void)ws_size;
    const float* pos = (const float*)d_in[0];
    const int n = in_sizes[0] / 3;           // 16384
    const int E = n * KNBR + n;              // 540672 edges incl. self loops

    // d_out layout (reference tuple flattened):
    //   [0, 2E)      edge_index (int32 bits)
    //   [2E, 3E)     edge_weight (f32)
    //   [3E, 6E)     edge_vec    (f32, [E][3])
    int*   ei = (int*)d_out;
    float* base = (float*)d_out;
    float* wt = base + (size_t)2 * E;
    float* vc = base + (size_t)3 * E;

    const int nblocks = n / 16;              // 1024 blocks x 4 wave32
    radius_graph_wmma_f32<<<nblocks, 128, 0, stream>>>(pos, n, ei, wt, vc, E);
}